// GCN_90744069030482
// MI455X (gfx1250) — compile-verified
//
#include <hip/hip_runtime.h>

// ---------------------------------------------------------------------------
// GCN (3x GCNConv + global_add_pool), fp32 end-to-end.
// GEMMs run on the CDNA5 WMMA pipe via V_WMMA_F32_16X16X4_F32 (exact fp32).
// Sparse aggregation is L2-resident atomic scatter-add (12.8MB << 192MB L2).
// ---------------------------------------------------------------------------

typedef __attribute__((ext_vector_type(2))) float v2f;
typedef __attribute__((ext_vector_type(8))) float v8f;

#define N_NODES 50000
#define N_EDGES 800000
#define CH 64
#define NUM_GRAPHS 256

// ---------------- degree / normalization ----------------

__global__ void gcn_deg_init(float* __restrict__ deg) {
    int i = blockIdx.x * blockDim.x + threadIdx.x;
    if (i < N_NODES) deg[i] = 1.0f;   // self loop
}

__global__ void gcn_deg_count(const int* __restrict__ dst, float* __restrict__ deg) {
    int e = blockIdx.x * blockDim.x + threadIdx.x;
    if (e < N_EDGES) atomicAdd(&deg[dst[e]], 1.0f);
}

__global__ void gcn_deg_to_dinv(float* __restrict__ deg) {
    int i = blockIdx.x * blockDim.x + threadIdx.x;
    if (i < N_NODES) deg[i] = rsqrtf(deg[i]);  // deg >= 1 always
}

// ---------------- dense GEMM: Y[M,64] = X[M,64] @ W[64,64] (WMMA fp32) -------
// One wave per 16x16 output tile. Block = 128 threads = 4 waves; wave w owns
// N-tile w (4 tiles cover 64 cols). blockIdx.x = M-tile (50000/16 = 3125).

__global__ void __launch_bounds__(128)
gcn_gemm_wmma(const float* __restrict__ X, const float* __restrict__ W,
              float* __restrict__ Y) {
    const int lane = threadIdx.x & 31;
    const int wave = threadIdx.x >> 5;
    const int m0 = blockIdx.x << 4;   // M-tile base
    const int n0 = wave << 4;         // N-tile base

    const int l15 = lane & 15;
    const int kb  = (lane >> 4) << 1; // lanes 0-15 -> K+0/K+1; lanes 16-31 -> K+2/K+3

    const float* __restrict__ xrow = X + (size_t)(m0 + l15) * CH;

    v8f c = {};
#pragma unroll
    for (int k0 = 0; k0 < CH; k0 += 4) {
        v2f a, b;
        // A fragment (16x4, M = l15): VGPR0 = K=kb, VGPR1 = K=kb+1
        a.x = xrow[k0 + kb];
        a.y = xrow[k0 + kb + 1];
        // B fragment (4x16, N = n0 + l15): mirrored K striping
        b.x = W[(size_t)(k0 + kb)     * CH + n0 + l15];
        b.y = W[(size_t)(k0 + kb + 1) * CH + n0 + l15];
        c = __builtin_amdgcn_wmma_f32_16x16x4_f32(
                /*neg_a=*/false, a, /*neg_b=*/false, b,
                /*c_mod=*/(short)0, c, /*reuse_a=*/false, /*reuse_b=*/false);
    }

    // C/D layout: VGPR r -> rows (r, r+8); lanes 16-31 are the +8 half.
    const int rowOff = (lane >> 4) << 3;
    const int col = n0 + l15;
#pragma unroll
    for (int r = 0; r < 8; ++r) {
        Y[(size_t)(m0 + rowOff + r) * CH + col] = c[r];
    }
}

// ---------------- aggregation ----------------
// out[i,c] = dinv[i]^2 * xw[i,c] + bias[c]   (self loop + bias)

__global__ void gcn_self_init(const float* __restrict__ xw,
                              const float* __restrict__ dinv,
                              const float* __restrict__ bias,
                              float* __restrict__ out) {
    int idx = blockIdx.x * blockDim.x + threadIdx.x;
    if (idx < N_NODES * CH) {
        int i = idx >> 6, cch = idx & 63;
        float di = dinv[i];
        out[idx] = di * di * xw[idx] + bias[cch];
    }
}

// out[dst,c] += dinv[src]*dinv[dst] * xw[src,c] ; 16 lanes/edge, float4/lane.

__global__ void gcn_edge_scatter(const int* __restrict__ src,
                                 const int* __restrict__ dst,
                                 const float* __restrict__ dinv,
                                 const float* __restrict__ xw,
                                 float* __restrict__ out) {
    int t = blockIdx.x * blockDim.x + threadIdx.x;
    int e = t >> 4;
    if (e >= N_EDGES) return;
    int lane = t & 15;
    int s = src[e], d = dst[e];
    float norm = dinv[s] * dinv[d];
    const float4* __restrict__ xs = (const float4*)(xw + (size_t)s * CH);
    float4 v = xs[lane];
    float* __restrict__ o = out + (size_t)d * CH + lane * 4;
    atomicAdd(o + 0, norm * v.x);
    atomicAdd(o + 1, norm * v.y);
    atomicAdd(o + 2, norm * v.z);
    atomicAdd(o + 3, norm * v.w);
}

__global__ void gcn_relu(float* __restrict__ h) {
    int idx = blockIdx.x * blockDim.x + threadIdx.x;
    if (idx < N_NODES * CH) h[idx] = fmaxf(h[idx], 0.0f);
}

// ---------------- global add pool ----------------

__global__ void gcn_zero_out(float* __restrict__ o) {
    int i = blockIdx.x * blockDim.x + threadIdx.x;
    if (i < NUM_GRAPHS * CH) o[i] = 0.0f;
}

__global__ void gcn_pool(const float* __restrict__ h,
                         const int* __restrict__ batch,
                         float* __restrict__ out) {
    int idx = blockIdx.x * blockDim.x + threadIdx.x;
    if (idx < N_NODES * CH) {
        int i = idx >> 6, cch = idx & 63;
        atomicAdd(&out[(size_t)batch[i] * CH + cch], h[idx]);
    }
}

// ---------------- host orchestration ----------------

extern "C" void kernel_launch(void* const* d_in, const int* in_sizes, int n_in,
                              void* d_out, int out_size, void* d_ws, size_t ws_size,
                              hipStream_t stream) {
    (void)in_sizes; (void)n_in; (void)out_size; (void)ws_size;

    const float* x     = (const float*)d_in[0];
    const int*   eidx  = (const int*)d_in[1];        // [2, E] flat
    const int*   batch = (const int*)d_in[2];
    const float* W1 = (const float*)d_in[3];
    const float* b1 = (const float*)d_in[4];
    const float* W2 = (const float*)d_in[5];
    const float* b2 = (const float*)d_in[6];
    const float* W3 = (const float*)d_in[7];
    const float* b3 = (const float*)d_in[8];
    float* out = (float*)d_out;

    const int* src = eidx;
    const int* dst = eidx + N_EDGES;

    // workspace layout (floats): [dinv: 50176] [bufA: N*64] [bufB: N*64]
    float* dinv = (float*)d_ws;
    float* bufA = dinv + 50176;
    float* bufB = bufA + (size_t)N_NODES * CH;

    const int B = 256;
    const int gNode  = (N_NODES + B - 1) / B;            // 196
    const int gEdge  = (N_EDGES + B - 1) / B;            // 3125
    const int gFeat  = (N_NODES * CH + B - 1) / B;       // 12500
    const int gScat  = (N_EDGES * 16 + B - 1) / B;       // 50000
    const int gGemm  = N_NODES / 16;                     // 3125 (exact)
    const int gPoolZ = (NUM_GRAPHS * CH + B - 1) / B;

    // degree -> dinv (shared by all 3 layers)
    gcn_deg_init   <<<gNode, B, 0, stream>>>(dinv);
    gcn_deg_count  <<<gEdge, B, 0, stream>>>(dst, dinv);
    gcn_deg_to_dinv<<<gNode, B, 0, stream>>>(dinv);

    // ---- layer 1 ----
    gcn_gemm_wmma   <<<gGemm, 128, 0, stream>>>(x, W1, bufA);
    gcn_self_init   <<<gFeat, B, 0, stream>>>(bufA, dinv, b1, bufB);
    gcn_edge_scatter<<<gScat, B, 0, stream>>>(src, dst, dinv, bufA, bufB);
    gcn_relu        <<<gFeat, B, 0, stream>>>(bufB);

    // ---- layer 2 ----
    gcn_gemm_wmma   <<<gGemm, 128, 0, stream>>>(bufB, W2, bufA);
    gcn_self_init   <<<gFeat, B, 0, stream>>>(bufA, dinv, b2, bufB);
    gcn_edge_scatter<<<gScat, B, 0, stream>>>(src, dst, dinv, bufA, bufB);
    gcn_relu        <<<gFeat, B, 0, stream>>>(bufB);

    // ---- layer 3 (no relu) ----
    gcn_gemm_wmma   <<<gGemm, 128, 0, stream>>>(bufB, W3, bufA);
    gcn_self_init   <<<gFeat, B, 0, stream>>>(bufA, dinv, b3, bufB);
    gcn_edge_scatter<<<gScat, B, 0, stream>>>(src, dst, dinv, bufA, bufB);

    // ---- global add pool ----
    gcn_zero_out<<<gPoolZ, B, 0, stream>>>(out);
    gcn_pool    <<<gFeat, B, 0, stream>>>(bufB, batch, out);
}